// BasicConvolutionBlock_24266565222402
// MI455X (gfx1250) — compile-verified
//
#include <hip/hip_runtime.h>
#include <hip/hip_bf16.h>

#define N_VOX   150000
#define KOFF    27
#define CIN     32
#define COUT    64
#define KC      (KOFF * CIN)          // 864
#define EPSBN   1e-5f
#define SLOPE   0.01f
#define NJOBS   ((N_VOX + 31) / 32)   // 4688 wave-jobs, 32 voxels each
#define WAVES_PER_BLK 4

// workspace layout (floats)
#define WS_ACC   0                    // 128 floats: per-channel sum, sumsq
#define WS_ZERO  128                  // 32 floats: zero block for masked rows
#define WS_WT    160                  // 864*64 floats: pair-interleaved W

typedef float v2f __attribute__((ext_vector_type(2)));
typedef float v4f __attribute__((ext_vector_type(4)));
typedef float v8f __attribute__((ext_vector_type(8)));

// ---------------------------------------------------------------------------
// Kernel 0: zero accumulators + zero-block, and build pair-interleaved W:
//   WtP[m][c][d][j] = W[2m + j][d*16 + c]   (m<432, c<16, d<4, j<2)
// One b128 from WtP + m*128 + c*8 yields B-fragments b0,b1 pre-packed in
// adjacent VGPR pairs; the next b128 (+4) yields b2,b3. No repack movs.
// ---------------------------------------------------------------------------
__global__ __launch_bounds__(256)
void prep_kernel(const float* __restrict__ W, float* __restrict__ ws) {
    const int t = blockIdx.x * 256 + threadIdx.x;
    if (t < 160) ws[t] = 0.0f;                 // accumulators + zero block
    if (t < KC * COUT) {
        const int r  = t >> 6;                 // W row (kc), 0..863
        const int ci = t & 63;                 // column within row
        const int d  = ci >> 4;                // 16-wide d-tile
        const int c  = ci & 15;                // column within d-tile
        const int m  = r >> 1;                 // row pair
        const int j  = r & 1;                  // row within pair
        ws[WS_WT + m * 128 + c * 8 + d * 2 + j] = W[t];
    }
}

// ---------------------------------------------------------------------------
// Kernel 1: gather-GEMM via V_WMMA_F32_16X16X4_F32.
// One wave -> 32 voxels x 64 channels (two M-tiles, 8 accumulators).
// Inner 4-wide kc step: 2x global_load_b64 (A) + 2x global_load_b128 (B)
// feeding 8 WMMAs; B fragments arrive pre-packed. Mask folded into a
// per-(row,k) pointer select to a zero block.
// ---------------------------------------------------------------------------
__global__ __launch_bounds__(WAVES_PER_BLK * 32)
void sparse_conv_wmma_kernel(const float* __restrict__ feats,
                             const int*   __restrict__ nbr,
                             const unsigned char* __restrict__ mask,
                             float* __restrict__ out,
                             float* __restrict__ ws) {
    __shared__ float s_sum[COUT];
    __shared__ float s_sq[COUT];

    const float* __restrict__ wt   = ws + WS_WT;
    const float* __restrict__ zbuf = ws + WS_ZERO;
    float*       __restrict__ gacc = ws + WS_ACC;

    const int tid  = threadIdx.x;
    if (tid < COUT) { s_sum[tid] = 0.0f; s_sq[tid] = 0.0f; }
    __syncthreads();

    const int lane = tid & 31;
    const int wv   = tid >> 5;
    const int job  = blockIdx.x * WAVES_PER_BLK + wv;
    const int hl   = lane >> 4;      // 0: lanes 0-15, 1: lanes 16-31
    const int col  = lane & 15;      // column within 16-wide tile / row id

    // Prefetch permuted W (216 KB = 1728 cache lines of 128 B).
    {
        const int t = blockIdx.x * (WAVES_PER_BLK * 32) + tid;
        if (t < (KC * COUT * 4) / 128)
            __builtin_prefetch((const char*)wt + (size_t)t * 128, 0, 2);
    }

    if (job < NJOBS) {
        const int  rb0 = job * 32;        // first 16-row tile
        const int  rb1 = rb0 + 16;        // second 16-row tile
        const bool v1  = (rb1 < N_VOX);   // tail guard (one wave only)

        v8f acc[8] = {};                  // [0..3]: tile0 d0..d3, [4..7]: tile1

        for (int k = 0; k < KOFF; ++k) {
            // one gather index + mask bit per (row, k); mask -> pointer select
            const int  ni0  = (rb0 + col) * KOFF + k;
            const int  idx0 = nbr[ni0];
            const bool mk0  = mask[ni0] != 0;
            const float* fp0 = mk0 ? (feats + (size_t)idx0 * CIN) : zbuf;

            const int  ni1  = (rb1 + col) * KOFF + k;
            const int  idx1 = v1 ? nbr[ni1] : 0;
            const bool mk1  = v1 && (mask[ni1] != 0);
            const float* fp1 = mk1 ? (feats + (size_t)idx1 * CIN) : zbuf;

            // base of pair-interleaved W for this k (lane-dependent half)
            // row pair m = k*16 + c0/2 + hl -> offset (c0/2)*128 floats
            const float* wbase = wt + (size_t)(k * 16 + hl) * 128 + col * 8;

            #pragma unroll
            for (int c0 = 0; c0 < CIN; c0 += 4) {
                // ---- A fragments: 16x4 f32 (2 VGPRs each) ----
                const int c = c0 + 2 * hl;
                const v2f a0 = *(const v2f*)(fp0 + c);   // 8B aligned
                const v2f a1 = *(const v2f*)(fp1 + c);

                // ---- B fragments: two b128, pre-packed pairs ----
                const float* wp = wbase + (size_t)(c0 >> 1) * 128;
                const v4f x = *(const v4f*)(wp);         // b0 | b1
                const v4f y = *(const v4f*)(wp + 4);     // b2 | b3
                const v2f b0 = __builtin_shufflevector(x, x, 0, 1);
                const v2f b1 = __builtin_shufflevector(x, x, 2, 3);
                const v2f b2 = __builtin_shufflevector(y, y, 0, 1);
                const v2f b3 = __builtin_shufflevector(y, y, 2, 3);

                acc[0] = __builtin_amdgcn_wmma_f32_16x16x4_f32(
                             false, a0, false, b0, (short)0, acc[0], false, false);
                acc[1] = __builtin_amdgcn_wmma_f32_16x16x4_f32(
                             false, a0, false, b1, (short)0, acc[1], false, false);
                acc[2] = __builtin_amdgcn_wmma_f32_16x16x4_f32(
                             false, a0, false, b2, (short)0, acc[2], false, false);
                acc[3] = __builtin_amdgcn_wmma_f32_16x16x4_f32(
                             false, a0, false, b3, (short)0, acc[3], false, false);
                acc[4] = __builtin_amdgcn_wmma_f32_16x16x4_f32(
                             false, a1, false, b0, (short)0, acc[4], false, false);
                acc[5] = __builtin_amdgcn_wmma_f32_16x16x4_f32(
                             false, a1, false, b1, (short)0, acc[5], false, false);
                acc[6] = __builtin_amdgcn_wmma_f32_16x16x4_f32(
                             false, a1, false, b2, (short)0, acc[6], false, false);
                acc[7] = __builtin_amdgcn_wmma_f32_16x16x4_f32(
                             false, a1, false, b3, (short)0, acc[7], false, false);
            }
        }

        // ---- write raw result + reduce per-channel sum / sumsq ----
        // C/D layout: VGPR j holds M=j (lanes 0-15) and M=8+j (lanes 16-31)
        #pragma unroll
        for (int d = 0; d < 4; ++d) {
            float s = 0.0f, q = 0.0f;
            #pragma unroll
            for (int j = 0; j < 8; ++j) {
                const float u = acc[d][j];
                const float w = acc[4 + d][j];
                out[(size_t)(rb0 + j + 8 * hl) * COUT + d * 16 + col] = u;
                if (v1)
                    out[(size_t)(rb1 + j + 8 * hl) * COUT + d * 16 + col] = w;
                s += u + w;            // w == 0 exactly when !v1
                q += u * u + w * w;
            }
            atomicAdd(&s_sum[d * 16 + col], s);   // ds_add_f32
            atomicAdd(&s_sq [d * 16 + col], q);
        }
    }

    __syncthreads();
    if (tid < COUT) {
        atomicAdd(&gacc[tid],        s_sum[tid]);  // global_atomic_add_f32
        atomicAdd(&gacc[COUT + tid], s_sq [tid]);
    }
}

// ---------------------------------------------------------------------------
// Kernel 2: BatchNorm (batch stats) + LeakyReLU, in-place on d_out.
// ---------------------------------------------------------------------------
__global__ __launch_bounds__(256)
void bn_lrelu_kernel(float* __restrict__ out,
                     const float* __restrict__ gacc,
                     const float* __restrict__ gamma,
                     const float* __restrict__ beta) {
    __shared__ float s_scale[COUT];
    __shared__ float s_bias[COUT];

    const int tid = threadIdx.x;
    if (tid < COUT) {
        const float inv_n = 1.0f / (float)N_VOX;
        const float mean  = gacc[tid] * inv_n;
        const float var   = gacc[COUT + tid] * inv_n - mean * mean;
        const float sc    = rsqrtf(var + EPSBN) * gamma[tid];
        s_scale[tid] = sc;
        s_bias[tid]  = beta[tid] - mean * sc;
    }
    __syncthreads();

    const size_t base = ((size_t)blockIdx.x * 256 + tid) * 4;
    if (base < (size_t)N_VOX * COUT) {
        float4 v = *(float4*)(out + base);
        const int ch = (int)(base & (COUT - 1));   // multiple of 4
        float r0 = v.x * s_scale[ch + 0] + s_bias[ch + 0];
        float r1 = v.y * s_scale[ch + 1] + s_bias[ch + 1];
        float r2 = v.z * s_scale[ch + 2] + s_bias[ch + 2];
        float r3 = v.w * s_scale[ch + 3] + s_bias[ch + 3];
        v.x = (r0 >= 0.0f) ? r0 : SLOPE * r0;
        v.y = (r1 >= 0.0f) ? r1 : SLOPE * r1;
        v.z = (r2 >= 0.0f) ? r2 : SLOPE * r2;
        v.w = (r3 >= 0.0f) ? r3 : SLOPE * r3;
        *(float4*)(out + base) = v;
    }
}

// ---------------------------------------------------------------------------
extern "C" void kernel_launch(void* const* d_in, const int* in_sizes, int n_in,
                              void* d_out, int out_size, void* d_ws, size_t ws_size,
                              hipStream_t stream) {
    const float* feats = (const float*)d_in[0];                // [N, 32]
    const float* W     = (const float*)d_in[1];                // [27, 32, 64]
    const float* gamma = (const float*)d_in[2];                // [64]
    const float* beta  = (const float*)d_in[3];                // [64]
    const int*   nbr   = (const int*)d_in[4];                  // [N, 27]
    const unsigned char* mask = (const unsigned char*)d_in[5]; // [N, 27] bool

    float* out = (float*)d_out;   // [N, 64]
    float* ws  = (float*)d_ws;    // accums + zero block + permuted W

    const int prep_blocks = (KC * COUT + 255) / 256;           // 216
    prep_kernel<<<prep_blocks, 256, 0, stream>>>(W, ws);

    const int nblocks = (NJOBS + WAVES_PER_BLK - 1) / WAVES_PER_BLK; // 1172
    sparse_conv_wmma_kernel<<<nblocks, WAVES_PER_BLK * 32, 0, stream>>>(
        feats, nbr, mask, out, ws);

    const size_t total_vec4 = (size_t)N_VOX * COUT / 4;        // 2.4M
    const int bn_blocks = (int)((total_vec4 + 255) / 256);     // 9375
    bn_lrelu_kernel<<<bn_blocks, 256, 0, stream>>>(out, ws + WS_ACC, gamma, beta);
}